// ChirpletKANLinear_60533269070048
// MI455X (gfx1250) — compile-verified
//
#include <hip/hip_runtime.h>
#include <hip/hip_bf16.h>
#include <math.h>

#define BDIM 1024
#define IN   512
#define OUT  512

typedef __attribute__((ext_vector_type(2))) float v2f;
typedef __attribute__((ext_vector_type(8))) float v8f;

// fast SiLU: x * 1/(1+exp(-x)); v_exp_f32 is exp2, fold log2(e)
__device__ __forceinline__ float silu_fast(float v) {
    float e = __expf(-v);                       // -> v_exp_f32 (+mul by log2e)
    return v * __builtin_amdgcn_rcpf(1.0f + e); // -> v_rcp_f32
}

// ---------------------------------------------------------------------------
// Kernel A: base_out = SiLU(x) @ Wb^T + bias   via V_WMMA_F32_16X16X4_F32
// One wave per 16x16 output tile. 8 waves per block.
// grid = (OUT/16, B/(16*8)), block = 256
// ---------------------------------------------------------------------------
__global__ __launch_bounds__(256) void base_gemm_kernel(
    const float* __restrict__ x, const float* __restrict__ bw,
    const float* __restrict__ bias, float* __restrict__ out) {

    const int lane  = threadIdx.x & 31;
    const int wv    = threadIdx.x >> 5;
    const int nbase = blockIdx.x * 16;
    const int mbase = (blockIdx.y * 8 + wv) * 16;

    const int r16  = lane & 15;   // A row / B col / D col
    const int half = lane >> 4;   // K-pair selector (0 -> K0,K1 ; 1 -> K2,K3)

    const float* xrow = x  + (size_t)(mbase + r16) * IN;
    const float* wrow = bw + (size_t)(nbase + r16) * IN;

    v8f c = {};
    for (int k = 0; k < IN; k += 4) {
        const int kb = k + half * 2;
        v2f a, b;
        a.x = silu_fast(xrow[kb]);
        a.y = silu_fast(xrow[kb + 1]);
        b.x = wrow[kb];
        b.y = wrow[kb + 1];
        // 8 args: (neg_a, A, neg_b, B, c_mod, C, reuse_a, reuse_b)
        c = __builtin_amdgcn_wmma_f32_16x16x4_f32(
                false, a, false, b, (short)0, c, false, false);
    }

    const float bv     = bias[nbase + r16];
    const int   rowoff = half * 8;  // D: lanes 0-15 hold M=r, lanes 16-31 hold M=r+8
#pragma unroll
    for (int r = 0; r < 8; ++r) {
        out[(size_t)(mbase + rowoff + r) * OUT + nbase + r16] = c[r] + bv;
    }
}

// ---------------------------------------------------------------------------
// Kernel B: out[b,o] += sum_i w[o,i] * cos(2*pi*f*(x-t)/s) * exp(-0.5*((x-t)/s)^2)
// Tile: 64 b x 32 o per block, IN chunked by 32. Derived params in LDS so the
// inner loop per element is 2 FMA + 3 MUL + v_cos_f32 + v_exp_f32 + 1 FMA.
// grid = (OUT/32, B/64), block = 256 (thread -> 1 o, 8 b's)
// ---------------------------------------------------------------------------
#define BM 64
#define BO 32
#define KC 32

__global__ __launch_bounds__(256) void chirplet_kernel(
    const float* __restrict__ x,  const float* __restrict__ cw,
    const float* __restrict__ sc, const float* __restrict__ tr,
    const float* __restrict__ fr, float* __restrict__ out) {

    __shared__ float xs    [BM][KC + 1];  // +1 pad: conflict-free stride-33
    __shared__ float p_invs[BO][KC + 1];
    __shared__ float p_nit [BO][KC + 1];
    __shared__ float p_fi  [BO][KC + 1];  // 2*pi*f/s
    __shared__ float p_nfit[BO][KC + 1];  // -2*pi*f*t/s
    __shared__ float p_w   [BO][KC + 1];

    const int t     = threadIdx.x;
    const int obase = blockIdx.x * BO;
    const int bbase = blockIdx.y * BM;
    const int o_l   = t & 31;   // lane-varying o -> coalesced out, conflict-free LDS
    const int brow  = t >> 5;   // 0..7

    const float TWO_PI = 6.283185307179586f;

    float acc[8];
#pragma unroll
    for (int j = 0; j < 8; ++j) acc[j] = 0.0f;

    for (int k0 = 0; k0 < IN; k0 += KC) {
        // ---- cooperative load: x tile (64x32, coalesced) ----
#pragma unroll
        for (int r = 0; r < (BM * KC) / 256; ++r) {
            int idx = t + 256 * r;
            int bl = idx >> 5, il = idx & 31;
            xs[bl][il] = x[(size_t)(bbase + bl) * IN + k0 + il];
        }
        // ---- cooperative load + transform params (32x32, coalesced) ----
#pragma unroll
        for (int r = 0; r < (BO * KC) / 256; ++r) {
            int idx = t + 256 * r;
            int ol = idx >> 5, il = idx & 31;
            size_t g = (size_t)(obase + ol) * IN + k0 + il;
            float s  = sc[g];
            float tt = tr[g];
            float f  = fr[g];
            float w  = cw[g];
            float invs = 1.0f / s;          // amortized: only 4M divisions total
            float fi   = TWO_PI * f * invs;
            p_invs[ol][il] = invs;
            p_nit [ol][il] = -tt * invs;
            p_fi  [ol][il] = fi;
            p_nfit[ol][il] = -fi * tt;
            p_w   [ol][il] = w;
        }
        __syncthreads();

        // ---- compute: each thread 8 (b,o) elements per i ----
#pragma unroll 4
        for (int i = 0; i < KC; ++i) {
            float invs = p_invs[o_l][i];
            float nit  = p_nit [o_l][i];
            float fi   = p_fi  [o_l][i];
            float nfit = p_nfit[o_l][i];
            float w    = p_w   [o_l][i];
#pragma unroll
            for (int j = 0; j < 8; ++j) {
                float xv = xs[brow + 8 * j][i];       // wave-uniform broadcast
                float z  = fmaf(xv, invs, nit);       // (x - t)/s
                float a  = fmaf(xv, fi, nfit);        // 2*pi*f*(x-t)/s  (radians)
                float cs = __cosf(a);                 // -> v_cos_f32
                float ex = __expf(-0.5f * (z * z));   // -> v_exp_f32
                acc[j] = fmaf(w, cs * ex, acc[j]);
            }
        }
        __syncthreads();
    }

#pragma unroll
    for (int j = 0; j < 8; ++j) {
        size_t o = (size_t)(bbase + brow + 8 * j) * OUT + obase + o_l;
        out[o] += acc[j];  // kernel A already wrote base+bias (same stream order)
    }
}

extern "C" void kernel_launch(void* const* d_in, const int* in_sizes, int n_in,
                              void* d_out, int out_size, void* d_ws, size_t ws_size,
                              hipStream_t stream) {
    const float* x  = (const float*)d_in[0];  // (B, IN)
    const float* bw = (const float*)d_in[1];  // base_weight (OUT, IN)
    const float* cw = (const float*)d_in[2];  // chirplet_weights (OUT, IN)
    const float* sc = (const float*)d_in[3];  // scale (OUT, IN)
    const float* tr = (const float*)d_in[4];  // translation (OUT, IN)
    const float* fr = (const float*)d_in[5];  // frequency (OUT, IN)
    const float* bi = (const float*)d_in[6];  // bias (OUT,)
    float* out = (float*)d_out;               // (B, OUT)

    dim3 gA(OUT / 16, BDIM / (16 * 8));       // (32, 8), 8 waves/block
    base_gemm_kernel<<<gA, 256, 0, stream>>>(x, bw, bi, out);

    dim3 gB(OUT / BO, BDIM / BM);             // (16, 16)
    chirplet_kernel<<<gB, 256, 0, stream>>>(x, cw, sc, tr, fr, out);
}